// WeightedMSELoss_64424509440434
// MI455X (gfx1250) — compile-verified
//
#include <hip/hip_runtime.h>
#include <hip/hip_bf16.h>

#define B_ 64
#define H_ 512
#define W_ 512
#define HW_ (H_ * W_)

typedef __attribute__((ext_vector_type(2))) float v2f;
typedef __attribute__((ext_vector_type(8))) float v8f;

// ---------------------------------------------------------------------------
// Kernel 1: Dsum[h,w] = sum_b (pred - targ)^2   (memory bound: 134 MB read)
// ---------------------------------------------------------------------------
__global__ __launch_bounds__(256) void dsum_kernel(const float* __restrict__ pred,
                                                   const float* __restrict__ targ,
                                                   float* __restrict__ Dsum) {
    const int idx = blockIdx.x * blockDim.x + threadIdx.x; // 0 .. HW-1
    float acc = 0.0f;
#pragma unroll 4
    for (int b = 0; b < B_; ++b) {
        if (b + 4 < B_) {
            __builtin_prefetch(pred + (size_t)(b + 4) * HW_ + idx, 0, 1);
            __builtin_prefetch(targ + (size_t)(b + 4) * HW_ + idx, 0, 1);
        }
        float d = pred[(size_t)b * HW_ + idx] - targ[(size_t)b * HW_ + idx];
        acc += d * d;
    }
    Dsum[idx] = acc;
}

// ---------------------------------------------------------------------------
// Kernel 2: colsum[b,w] = sum_h targ[b,h,w]   (PDDs * H); coalesced over w
// ---------------------------------------------------------------------------
__global__ __launch_bounds__(256) void colsum_kernel(const float* __restrict__ targ,
                                                     float* __restrict__ colsum) {
    const int idx = blockIdx.x * blockDim.x + threadIdx.x; // 0 .. B*W-1
    const int b = idx >> 9;
    const int w = idx & (W_ - 1);
    const float* p = targ + (size_t)b * HW_ + w;
    float acc = 0.0f;
    for (int h = 0; h < H_; ++h) acc += p[(size_t)h * W_];
    colsum[idx] = acc;
}

// ---------------------------------------------------------------------------
// Kernel 3: rowsum[b,h] = sum_w targ[b,h,w]  (profiles * W); wave per row
// ---------------------------------------------------------------------------
__global__ __launch_bounds__(256) void rowsum_kernel(const float* __restrict__ targ,
                                                     float* __restrict__ rowsum) {
    const int wave = (blockIdx.x * blockDim.x + threadIdx.x) >> 5; // 0 .. B*H-1
    const int lane = threadIdx.x & 31;
    const float* p = targ + (size_t)wave * W_;
    float acc = 0.0f;
    for (int w = lane; w < W_; w += 32) acc += p[w];
#pragma unroll
    for (int off = 16; off > 0; off >>= 1) acc += __shfl_down(acc, off, 32);
    if (lane == 0) rowsum[wave] = acc;
}

// ---------------------------------------------------------------------------
// Kernel 4: per-batch parameters -> separable Gaussian factors gx[b,w], gy[b,h]
// One block per batch, 512 threads (one per h and per w).
// ---------------------------------------------------------------------------
__global__ __launch_bounds__(512) void params_kernel(const float* __restrict__ targ,
                                                     const float* __restrict__ rowsum,
                                                     const float* __restrict__ colsum,
                                                     float* __restrict__ gx,
                                                     float* __restrict__ gy) {
    __shared__ float prof[H_];
    __shared__ float pdds[W_];
    __shared__ float rv[512];
    __shared__ int ri[512];
    __shared__ float s_q51, s_q52, s_maxv, s_pmax;
    __shared__ int s_jmax, s_depth, s_lat, s_pmaxidx, s_lower, s_upper;

    const int t = threadIdx.x;
    const int b = blockIdx.x;

    prof[t] = rowsum[b * H_ + t] * (1.0f / W_);
    pdds[t] = colsum[b * W_ + t] * (1.0f / H_);
    __syncthreads();

    // --- quantile(0.1) over profiles: linear interp between sorted[51],[52]
    {
        float v = prof[t];
        int r = 0;
        for (int j = 0; j < 512; ++j) {
            float u = prof[j];
            r += (u < v) || (u == v && j < t); // stable rank (bijection)
        }
        if (r == 51) s_q51 = v;
        if (r == 52) s_q52 = v;
    }
    __syncthreads();
    {
        float q = 0.9f * s_q51 + 0.1f * s_q52; // pos = 0.1*(512-1) = 51.1
        float p = prof[t] - q;
        __syncthreads();
        prof[t] = p;
    }
    __syncthreads();

    // --- jmax = argmax(PDDs) (first occurrence), maxv
    rv[t] = pdds[t]; ri[t] = t;
    __syncthreads();
    for (int s = 256; s > 0; s >>= 1) {
        if (t < s) {
            if (rv[t + s] > rv[t] || (rv[t + s] == rv[t] && ri[t + s] < ri[t])) {
                rv[t] = rv[t + s]; ri[t] = ri[t + s];
            }
        }
        __syncthreads();
    }
    if (t == 0) { s_jmax = ri[0]; s_maxv = rv[0]; }
    __syncthreads();

    // --- depth_idx = argmin |0.8*maxv - PDDs| with w < jmax masked to +inf
    {
        float ad = (t < s_jmax) ? __builtin_inff() : fabsf(0.8f * s_maxv - pdds[t]);
        rv[t] = ad; ri[t] = t;
    }
    __syncthreads();
    for (int s = 256; s > 0; s >>= 1) {
        if (t < s) {
            if (rv[t + s] < rv[t] || (rv[t + s] == rv[t] && ri[t + s] < ri[t])) {
                rv[t] = rv[t + s]; ri[t] = ri[t + s];
            }
        }
        __syncthreads();
    }
    if (t == 0) s_depth = ri[0];
    __syncthreads();

    // --- lateral_idx = argmax over h of target[b,0,h,depth_idx]
    rv[t] = targ[((size_t)b * H_ + t) * W_ + s_depth]; ri[t] = t;
    __syncthreads();
    for (int s = 256; s > 0; s >>= 1) {
        if (t < s) {
            if (rv[t + s] > rv[t] || (rv[t + s] == rv[t] && ri[t + s] < ri[t])) {
                rv[t] = rv[t + s]; ri[t] = ri[t + s];
            }
        }
        __syncthreads();
    }
    if (t == 0) s_lat = ri[0];
    __syncthreads();

    // --- FWHM of quantile-shifted profiles
    rv[t] = prof[t]; ri[t] = t;
    __syncthreads();
    for (int s = 256; s > 0; s >>= 1) {
        if (t < s) {
            if (rv[t + s] > rv[t] || (rv[t + s] == rv[t] && ri[t + s] < ri[t])) {
                rv[t] = rv[t + s]; ri[t] = ri[t + s];
            }
        }
        __syncthreads();
    }
    if (t == 0) { s_pmax = rv[0]; s_pmaxidx = ri[0]; }
    __syncthreads();

    const float half = 0.5f * s_pmax;
    const bool below = prof[t] < half;
    ri[t] = (below && t < s_pmaxidx) ? t : -1; // lower = max
    __syncthreads();
    for (int s = 256; s > 0; s >>= 1) {
        if (t < s) { if (ri[t + s] > ri[t]) ri[t] = ri[t + s]; }
        __syncthreads();
    }
    if (t == 0) s_lower = ri[0];
    __syncthreads();
    ri[t] = (below && t >= s_pmaxidx) ? t : H_; // upper = min
    __syncthreads();
    for (int s = 256; s > 0; s >>= 1) {
        if (t < s) { if (ri[t + s] < ri[t]) ri[t] = ri[t + s]; }
        __syncthreads();
    }
    if (t == 0) s_upper = ri[0];
    __syncthreads();

    const int fwhm_i = (s_lower >= 0 && s_upper < H_) ? (s_upper - s_lower) : -1;

    // --- Gaussian factors (separable form of reference weights)
    const float d = (float)s_depth;
    const float lat = (float)s_lat;
    const float fw = (float)fwhm_i;
    const float twoSd2 = (0.1f * d - d) * (0.1f * d - d) / logf(10.0f); // 2*sigma_depth^2
    const float Kl = (3.0f / 2.355f) * fw;
    const float twoSl2 = (Kl - lat) * (Kl - lat) / logf(20.0f);         // 2*sigma_lat^2
    const float tf = (float)t;
    gx[b * W_ + t] = __expf(-((tf - d) * (tf - d)) / twoSd2) * 0.0f +
                     expf(-((tf - d) * (tf - d)) / twoSd2);
    gy[b * H_ + t] = expf(-((tf - lat) * (tf - lat)) / twoSl2);
}

// ---------------------------------------------------------------------------
// Kernel 5: WMMA contraction.  M = Dsum(512x512) @ Gx^T(512x64) tile-by-tile,
// fused with gy dot-product. One wave per 16(h) x 16(b) tile -> 128 waves.
// V_WMMA_F32_16X16X4_F32, K-loop over W in steps of 4.
//   A fragment (16x4 f32): lanes 0-15 hold K={0,1}, lanes 16-31 K={2,3}
//   B fragment (4x16 f32): mirrored; B[k][n] = gx[bbase+n][w=k]
//   C/D: vgpr r -> M = r + 8*(lane>=16), N = lane&15
// ---------------------------------------------------------------------------
__global__ __launch_bounds__(128) void wmma_contract_kernel(
    const float* __restrict__ Dsum, const float* __restrict__ gx,
    const float* __restrict__ gy, float* __restrict__ partials) {
    const int wave = (blockIdx.x * blockDim.x + threadIdx.x) >> 5; // 0..127
    const int lane = threadIdx.x & 31;
    const int tileH = wave & 31;  // 32 h-tiles
    const int tileB = wave >> 5;  // 4  b-tiles
    const int hbase = tileH * 16;
    const int bbase = tileB * 16;
    const int m = lane & 15;
    const int kh = lane >> 4; // which K-pair this half-wave holds

    const float* arow = Dsum + (size_t)(hbase + m) * W_ + kh * 2;
    const float* brow = gx + (size_t)(bbase + m) * W_ + kh * 2;

    v8f acc = {};
#pragma unroll 4
    for (int k = 0; k < W_; k += 4) {
        v2f a = *(const v2f*)(arow + k); // 8B-aligned: (k + 2*kh) even
        v2f bf = *(const v2f*)(brow + k);
        acc = __builtin_amdgcn_wmma_f32_16x16x4_f32(
            /*neg_a=*/false, a, /*neg_b=*/false, bf,
            /*c_mod=*/(short)0, acc, /*reuse_a=*/false, /*reuse_b=*/false);
    }

    // Fuse: local += M[h, bcol] * gy[bcol, h]
    float local = 0.0f;
    const int bcol = bbase + m;
#pragma unroll
    for (int r = 0; r < 8; ++r) {
        const int h = hbase + 8 * kh + r;
        local += acc[r] * gy[bcol * H_ + h];
    }
#pragma unroll
    for (int off = 16; off > 0; off >>= 1) local += __shfl_down(local, off, 32);
    if (lane == 0) partials[wave] = local;
}

// ---------------------------------------------------------------------------
// Kernel 6: deterministic fixed-order finalize
// ---------------------------------------------------------------------------
__global__ void finalize_kernel(const float* __restrict__ partials,
                                float* __restrict__ out) {
    if (threadIdx.x == 0 && blockIdx.x == 0) {
        float s = 0.0f;
        for (int i = 0; i < 128; ++i) s += partials[i];
        out[0] = s * (1.0f / 1073741824.0f); // / (B*B*H*W) = 2^30
    }
}

extern "C" void kernel_launch(void* const* d_in, const int* in_sizes, int n_in,
                              void* d_out, int out_size, void* d_ws, size_t ws_size,
                              hipStream_t stream) {
    const float* pred = (const float*)d_in[0];
    const float* targ = (const float*)d_in[1];
    float* out = (float*)d_out;

    // Workspace layout (floats): Dsum | colsum | rowsum | gx | gy | partials
    float* Dsum = (float*)d_ws;          // H*W        = 262144
    float* colsum = Dsum + HW_;          // B*W        =  32768
    float* rowsum = colsum + B_ * W_;    // B*H        =  32768
    float* gx = rowsum + B_ * H_;        // B*W        =  32768
    float* gy = gx + B_ * W_;            // B*H        =  32768
    float* partials = gy + B_ * H_;      // 128

    dsum_kernel<<<HW_ / 256, 256, 0, stream>>>(pred, targ, Dsum);
    colsum_kernel<<<(B_ * W_) / 256, 256, 0, stream>>>(targ, colsum);
    rowsum_kernel<<<(B_ * H_) / 8, 256, 0, stream>>>(targ, rowsum); // 8 waves/blk
    params_kernel<<<B_, 512, 0, stream>>>(targ, rowsum, colsum, gx, gy);
    wmma_contract_kernel<<<32, 128, 0, stream>>>(Dsum, gx, gy, partials);
    finalize_kernel<<<1, 32, 0, stream>>>(partials, out);
}